// VEMG2PoseWithInitialState_18726057411419
// MI455X (gfx1250) — compile-verified
//
#include <hip/hip_runtime.h>
#include <hip/hip_bf16.h>
#include <math.h>

typedef __attribute__((ext_vector_type(2))) float v2f;
typedef __attribute__((ext_vector_type(8))) float v8f;

__device__ __forceinline__ v8f wmma_f32(v2f a, v2f b, v8f c) {
  // D = A(16x4 f32) * B(4x16 f32) + C(16x16 f32)
  return __builtin_amdgcn_wmma_f32_16x16x4_f32(false, a, false, b, (short)0, c,
                                               false, false);
}

// ---------------------------------------------------------------------------
// Problem constants (from reference)
// ---------------------------------------------------------------------------
#define B_      32
#define C_IN    16
#define T_IN    8000
#define F_      512
#define K_      65
#define CK_     (C_IN * K_)      // 1040
#define H_      512
#define OUTCH   20
#define IN1     (F_ + OUTCH)     // 532
#define NTIME   198              // round(((8000-64)/2000)*50)
#define NPOS    50               // round(2000*50/2000)
#define NT_OUT  7936             // T - 64
#define CONV_L  7936             // valid conv length
#define N2      (2 * OUTCH)      // 40

// ---------------------------------------------------------------------------
// Prep kernels: weight transpose / WMMA-native k-group-of-4 packing
// ---------------------------------------------------------------------------
__global__ void transpose_w(const float* __restrict__ cw, float* __restrict__ Wt) {
  int idx = blockIdx.x * 256 + threadIdx.x;          // over 512*1040
  if (idx < F_ * CK_) {
    int f = idx / CK_, ck = idx - f * CK_;
    Wt[(size_t)ck * F_ + f] = cw[idx];
  }
}

// w1p[(k/4)*512 + n][k%4] = w1[k][n]  -> B-frag becomes one b64 load
__global__ void pack_w1(const float* __restrict__ w1, float* __restrict__ w1p) {
  int idx = blockIdx.x * 256 + threadIdx.x;          // over 532*512
  if (idx < IN1 * H_) {
    int k = idx / H_, n = idx - k * H_;
    w1p[((size_t)(k >> 2) * H_ + n) * 4 + (k & 3)] = w1[idx];
  }
}

__global__ void pack_w2(const float* __restrict__ w2, float* __restrict__ w2p) {
  int idx = blockIdx.x * 256 + threadIdx.x;          // over 512*40
  if (idx < H_ * N2) {
    int k = idx / N2, n = idx - k * N2;
    w2p[((size_t)(k >> 2) * N2 + n) * 4 + (k & 3)] = w2[idx];
  }
}

// ---------------------------------------------------------------------------
// Conv-as-GEMM: feats[t][b][f] = sum_{c,k} x_interp[t,b,c,k] * conv_w[f,c,k]
// (interp folded into the A gather; only the 198 used positions computed)
// One block = 32(M=batch, fixed t) x 64(N=f) tile, 8 waves, LDS-staged K.
// ---------------------------------------------------------------------------
__global__ void conv_gemm(const float* __restrict__ emg,
                          const float* __restrict__ Wt,
                          const float* __restrict__ conv_b,
                          float* __restrict__ feats) {
  const int t  = blockIdx.x;           // 0..197  (time step == M tile of 32)
  const int n0 = blockIdx.y * 64;      // feature block

  // align_corners=True source position (float32 math, matches reference)
  float src = (float)t * (7935.0f / 197.0f);
  int lo = (int)floorf(src);
  if (lo > CONV_L - 2) lo = CONV_L - 2;
  float w = src - (float)lo;

  __shared__ float lA[32][68];
  __shared__ float lB[64][68];

  const int tid   = threadIdx.x;       // 0..255
  const int lane  = tid & 31;
  const int wave  = tid >> 5;          // 0..7 -> 2x4 tiles
  const int wm    = wave >> 2;         // 0..1
  const int wn    = wave & 3;          // 0..3
  const int l15   = lane & 15;
  const int khalf = 2 * (lane >> 4);   // A/B K sub-lane per ISA layout

  v8f acc = {};
  for (int k0 = 0; k0 < CK_; k0 += 64) {
    int klen = CK_ - k0; if (klen > 64) klen = 64;   // last chunk = 16
    // A tile: interpolated emg window gather, 32 x klen
    for (int idx = tid; idx < 32 * 64; idx += 256) {
      int m = idx >> 6, kk = idx & 63;
      float v = 0.0f;
      if (kk < klen) {
        int ck = k0 + kk;
        int c = ck / K_, k = ck - c * K_;
        const float* p = emg + ((size_t)(m * C_IN + c)) * T_IN + lo + k;
        v = p[0] * (1.0f - w) + p[1] * w;
      }
      lA[m][kk] = v;
    }
    // B tile: klen x 64 of Wt
    for (int idx = tid; idx < 64 * 64; idx += 256) {
      int kk = idx >> 6, n = idx & 63;
      float v = 0.0f;
      if (kk < klen) v = Wt[(size_t)(k0 + kk) * F_ + n0 + n];
      lB[kk][n] = v;
    }
    __syncthreads();
    const int mrow = wm * 16 + l15;
    const int ncol = wn * 16 + l15;
    for (int kk = 0; kk < klen; kk += 4) {
      v2f a, b;
      a.x = lA[mrow][kk + khalf];
      a.y = lA[mrow][kk + khalf + 1];
      b.x = lB[kk + khalf][ncol];
      b.y = lB[kk + khalf + 1][ncol];
      acc = wmma_f32(a, b, acc);
    }
    __syncthreads();
  }
  // store C (layout: VGPR i -> M = i + 8*(lane>>4), N = lane&15)
  const int nglob = n0 + wn * 16 + l15;
  const float bias = conv_b[nglob];
  const int mbase = wm * 16 + 8 * (lane >> 4);
  for (int i = 0; i < 8; ++i) {
    feats[((size_t)(t * 32 + mbase + i)) * F_ + nglob] = acc[i] + bias;
  }
}

// ---------------------------------------------------------------------------
// Sequential scan: single persistent workgroup (32 waves on one WGP).
//   h = relu([feat_t | prev] @ w1 + b1) ; out = h @ w2 + b2
//   pred = (t < 50) ? pos : prev + vel
// feat_t staged to LDS once per step (coalesced b128); W fragments are
// single b64 loads from the packed layouts. ~141 KB dynamic LDS (<320 KB).
// ---------------------------------------------------------------------------
__global__ __launch_bounds__(1024) void scan_kernel(
    const float* __restrict__ feats, const float* __restrict__ ja,
    const int* __restrict__ provide, const float* __restrict__ w1p,
    const float* __restrict__ b1, const float* __restrict__ w2p,
    const float* __restrict__ b2, float* __restrict__ preds) {
  extern __shared__ float smem[];
  float (*prev)[OUTCH] = (float(*)[OUTCH])smem;                      // 32x20
  float (*fbuf)[516]   = (float(*)[516])(smem + 640);                // 32x516
  float (*hbuf)[516]   = (float(*)[516])(smem + 640 + 16512);        // 32x516
  float (*outb)[48]    = (float(*)[48])(smem + 640 + 2 * 16512);     // 32x48

  const int tid   = threadIdx.x;
  const int lane  = tid & 31;
  const int wave  = tid >> 5;          // 0..31
  const int l15   = lane & 15;
  const int khalf = 2 * (lane >> 4);

  // initial_pos = joint_angles[:, :, 32]  (or zeros)
  const int pv = provide[0];
  if (tid < B_ * OUTCH) {
    int b = tid / OUTCH, j = tid - b * OUTCH;
    prev[b][j] = pv ? ja[((size_t)(b * OUTCH + j)) * T_IN + 32] : 0.0f;
  }
  __syncthreads();

  for (int t = 0; t < NTIME; ++t) {
    // ---- stage feat_t[32][512] into LDS, coalesced b128 ----
    {
      const float4* src = (const float4*)(feats + (size_t)t * B_ * F_);
      for (int i = tid; i < 32 * 128; i += 1024) {
        int m = i >> 7, q = i & 127;     // row, float4 within row
        float4 v = src[i];
        fbuf[m][q * 4 + 0] = v.x;
        fbuf[m][q * 4 + 1] = v.y;
        fbuf[m][q * 4 + 2] = v.z;
        fbuf[m][q * 4 + 3] = v.w;
      }
    }
    __syncthreads();
    // ---- GEMM1: h[32,512] = relu(X[32,532] @ w1 + b1) ----
    {
      const int ncol = wave * 16 + l15;            // wave owns one N-tile col
      v8f c0 = {}, c1 = {};
      for (int kk = 0; kk < IN1; kk += 4) {
        const int g = kk >> 2;
        const int kA = kk + khalf;
        v2f b = *(const v2f*)(w1p + ((size_t)g * H_ + ncol) * 4 + khalf);
        v2f a0, a1;
        if (kA < F_) {                             // feature part of X
          a0 = *(const v2f*)&fbuf[l15][kA];
          a1 = *(const v2f*)&fbuf[16 + l15][kA];
        } else {                                   // carried state part of X
          a0 = *(const v2f*)&prev[l15][kA - F_];
          a1 = *(const v2f*)&prev[16 + l15][kA - F_];
        }
        c0 = wmma_f32(a0, b, c0);
        c1 = wmma_f32(a1, b, c1);
      }
      const float bn = b1[ncol];
      const int mb = 8 * (lane >> 4);
      for (int i = 0; i < 8; ++i) {
        hbuf[mb + i][ncol]      = fmaxf(c0[i] + bn, 0.0f);
        hbuf[16 + mb + i][ncol] = fmaxf(c1[i] + bn, 0.0f);
      }
    }
    __syncthreads();
    // ---- GEMM2: out[32,40] = h @ w2 + b2  (2x3 tiles on waves 0..5) ----
    if (wave < 6) {
      const int wm = wave / 3, wnt = wave - wm * 3;
      const int ncol = wnt * 16 + l15;
      const bool nok = ncol < N2;
      v8f c = {};
      for (int kk = 0; kk < H_; kk += 4) {
        const int g = kk >> 2;
        const int kA = kk + khalf;
        v2f a = *(const v2f*)&hbuf[wm * 16 + l15][kA];
        v2f b;
        if (nok) {
          b = *(const v2f*)(w2p + ((size_t)g * N2 + ncol) * 4 + khalf);
        } else {
          b.x = 0.0f; b.y = 0.0f;
        }
        c = wmma_f32(a, b, c);
      }
      if (nok) {
        const float bn = b2[ncol];
        const int mb = wm * 16 + 8 * (lane >> 4);
        for (int i = 0; i < 8; ++i) outb[mb + i][ncol] = c[i] + bn;
      }
    }
    __syncthreads();
    // ---- carry update + emit pred[t] ----
    if (tid < B_ * OUTCH) {
      int b = tid / OUTCH, j = tid - b * OUTCH;
      float pos = outb[b][j], vel = outb[b][j + OUTCH];
      float p = prev[b][j];
      float pr = (t < NPOS) ? pos : p + vel;
      prev[b][j] = pr;
      preds[((size_t)t * B_ + b) * OUTCH + j] = pr;
    }
    __syncthreads();
  }
}

// ---------------------------------------------------------------------------
// Finalize: out = concat(pred_up[32,20,7936], ja[32,20,7936], mask[32,7936])
// pred_up = interp(pred, align_corners=False)
// ---------------------------------------------------------------------------
__global__ void finalize(const float* __restrict__ preds,
                         const float* __restrict__ ja,
                         const unsigned char* __restrict__ mask_in,
                         float* __restrict__ out) {
  const size_t n1 = (size_t)B_ * OUTCH * NT_OUT;   // pred_up
  const size_t n2 = n1;                            // ja
  const size_t n3 = (size_t)B_ * NT_OUT;           // mask
  size_t idx = (size_t)blockIdx.x * 256 + threadIdx.x;
  if (idx < n1) {
    int i  = (int)(idx % NT_OUT);
    int bj = (int)(idx / NT_OUT);
    int j = bj % OUTCH, b = bj / OUTCH;
    float src = ((float)i + 0.5f) * ((float)NTIME / (float)NT_OUT) - 0.5f;
    src = fminf(fmaxf(src, 0.0f), (float)(NTIME - 1));
    int lo = (int)floorf(src);
    int hi = lo + 1; if (hi > NTIME - 1) hi = NTIME - 1;
    float w = src - (float)lo;
    float v0 = preds[((size_t)lo * B_ + b) * OUTCH + j];
    float v1 = preds[((size_t)hi * B_ + b) * OUTCH + j];
    out[idx] = v0 * (1.0f - w) + v1 * w;
  } else if (idx < n1 + n2) {
    size_t r = idx - n1;
    int i = (int)(r % NT_OUT);
    int bj = (int)(r / NT_OUT);
    out[idx] = ja[(size_t)bj * T_IN + 32 + i];
  } else if (idx < n1 + n2 + n3) {
    size_t r = idx - n1 - n2;
    int i = (int)(r % NT_OUT);
    int b = (int)(r / NT_OUT);
    out[idx] = mask_in[(size_t)b * T_IN + 32 + i] ? 1.0f : 0.0f;
  }
}

// ---------------------------------------------------------------------------
extern "C" void kernel_launch(void* const* d_in, const int* in_sizes, int n_in,
                              void* d_out, int out_size, void* d_ws, size_t ws_size,
                              hipStream_t stream) {
  const float* emg            = (const float*)d_in[0];
  const float* ja             = (const float*)d_in[1];
  const unsigned char* maskin = (const unsigned char*)d_in[2];
  const int* provide          = (const int*)d_in[3];
  const float* conv_w         = (const float*)d_in[4];
  const float* conv_b         = (const float*)d_in[5];
  const float* w1             = (const float*)d_in[6];
  const float* b1             = (const float*)d_in[7];
  const float* w2             = (const float*)d_in[8];
  const float* b2             = (const float*)d_in[9];

  float* ws    = (float*)d_ws;
  float* Wt    = ws;                               // 1040*512    = 532480 f
  float* feats = Wt + (size_t)CK_ * F_;            // 198*32*512  = 3244032 f
  float* preds = feats + (size_t)NTIME * B_ * F_;  // 198*32*20   = 126720 f
  float* w1p   = preds + (size_t)NTIME * B_ * OUTCH; // 532*512   = 272384 f
  float* w2p   = w1p + (size_t)IN1 * H_;           // 512*40      = 20480 f

  transpose_w<<<(F_ * CK_ + 255) / 256, 256, 0, stream>>>(conv_w, Wt);
  pack_w1<<<(IN1 * H_ + 255) / 256, 256, 0, stream>>>(w1, w1p);
  pack_w2<<<(H_ * N2 + 255) / 256, 256, 0, stream>>>(w2, w2p);

  dim3 g2(NTIME, F_ / 64);
  conv_gemm<<<g2, 256, 0, stream>>>(emg, Wt, conv_b, feats);

  size_t shmem = (size_t)(640 + 2 * 32 * 516 + 32 * 48) * sizeof(float);
  scan_kernel<<<1, 1024, shmem, stream>>>(feats, ja, provide, w1p, b1, w2p, b2,
                                          preds);

  size_t total = (size_t)2 * B_ * OUTCH * NT_OUT + (size_t)B_ * NT_OUT;
  finalize<<<(unsigned)((total + 255) / 256), 256, 0, stream>>>(preds, ja, maskin,
                                                               (float*)d_out);
}